// loss_function_82171314307686
// MI455X (gfx1250) — compile-verified
//
#include <hip/hip_runtime.h>
#include <math.h>

// ---------------------------------------------------------------------------
// loss = -mean(log(max(label ? p : 1-p, 1e-8))) + 0.002*(||W1||_F + ||W2||_F)
//
// Memory-bound streaming reduction (~168 MB -> ~7 us floor @ 23.3 TB/s).
// - b128 non-temporal loads (single-use streams, keep L2 clean)
// - per-block reduction via V_WMMA_F32_16X16X4_F32 with all-ones B
//   (f32 matrix unit used as a 64-way cross-lane adder; exact in f32)
// - deterministic two-pass reduction (no float atomics), f64 finalize
// ---------------------------------------------------------------------------

typedef __attribute__((ext_vector_type(2))) float v2f;
typedef __attribute__((ext_vector_type(8))) float v8f;
typedef __attribute__((ext_vector_type(4))) float f32x4;
typedef __attribute__((ext_vector_type(4))) int   i32x4;

#define THREADS     256
#define NLL_BLOCKS  2048
#define W1_BLOCKS   512
#define W2_BLOCKS   512
#define TOTAL_BLOCKS (NLL_BLOCKS + W1_BLOCKS + W2_BLOCKS)
#define NEAR_0      1e-8f

// Block-wide sum of 256 f32 values. Wave 0 folds 256 -> 64 partials (2 per
// lane), then one V_WMMA_F32_16X16X4_F32 with B = ones produces row sums in
// every column of D; summing column `lane` across lanes l and l^16 gives the
// grand total. Valid result in thread 0 (actually in all of wave 0).
__device__ __forceinline__ float block_sum_wmma(float v) {
    __shared__ float lds[THREADS];
    const int tid = threadIdx.x;
    lds[tid] = v;
    __syncthreads();
    float out = 0.0f;
    if (tid < 32) {                    // one full wave -> EXEC all ones
        const int i0 = 2 * tid;
        const int i1 = 2 * tid + 1;
        float a0 = lds[i0] + lds[i0 + 64] + lds[i0 + 128] + lds[i0 + 192];
        float a1 = lds[i1] + lds[i1 + 64] + lds[i1 + 128] + lds[i1 + 192];
        v2f A = {a0, a1};              // 64 partials striped as 16x4 A-matrix
        v2f B = {1.0f, 1.0f};          // all-ones 4x16 -> D[m][n] = rowsum(A,m)
        v8f C = {};                    // zero accumulator
        v8f D = __builtin_amdgcn_wmma_f32_16x16x4_f32(
            /*neg_a=*/false, A, /*neg_b=*/false, B,
            /*c_mod=*/(short)0, C, /*reuse_a=*/false, /*reuse_b=*/false);
        float t = D[0] + D[1] + D[2] + D[3] + D[4] + D[5] + D[6] + D[7];
        out = t + __shfl_xor(t, 16, 32);   // lanes l, l^16 hold M=0..7 / 8..15
    }
    return out;
}

__device__ __forceinline__ float ssq_partial(const float* __restrict__ W,
                                             int n, int gtid, int nthr) {
    float acc = 0.0f;
    const int n4 = n >> 2;
    const f32x4* __restrict__ w4 = reinterpret_cast<const f32x4*>(W);
    for (int i = gtid; i < n4; i += nthr) {
        f32x4 wv = __builtin_nontemporal_load(w4 + i);
        acc += wv.x * wv.x;
        acc += wv.y * wv.y;
        acc += wv.z * wv.z;
        acc += wv.w * wv.w;
    }
    for (int i = (n4 << 2) + gtid; i < n; i += nthr) {  // tail (unused: n%4==0)
        float w = W[i];
        acc += w * w;
    }
    return acc;
}

__global__ __launch_bounds__(THREADS) void partials_kernel(
    const float* __restrict__ p, const int* __restrict__ lab,
    const float* __restrict__ W1, const float* __restrict__ W2,
    float* __restrict__ partials, int n, int nw1, int nw2)
{
    const int bid = blockIdx.x;
    float acc = 0.0f;

    if (bid < NLL_BLOCKS) {
        const int gtid = bid * THREADS + threadIdx.x;
        const int nthr = NLL_BLOCKS * THREADS;
        const int n4 = n >> 2;
        const f32x4* __restrict__ p4 = reinterpret_cast<const f32x4*>(p);
        const i32x4* __restrict__ l4 = reinterpret_cast<const i32x4*>(lab);
        for (int i = gtid; i < n4; i += nthr) {
            f32x4 pv = __builtin_nontemporal_load(p4 + i);
            i32x4 lv = __builtin_nontemporal_load(l4 + i);
            float s0 = lv.x ? pv.x : 1.0f - pv.x;
            float s1 = lv.y ? pv.y : 1.0f - pv.y;
            float s2 = lv.z ? pv.z : 1.0f - pv.z;
            float s3 = lv.w ? pv.w : 1.0f - pv.w;
            acc += __logf(fmaxf(s0, NEAR_0));
            acc += __logf(fmaxf(s1, NEAR_0));
            acc += __logf(fmaxf(s2, NEAR_0));
            acc += __logf(fmaxf(s3, NEAR_0));
        }
        for (int i = (n4 << 2) + gtid; i < n; i += nthr) {  // tail
            float s = lab[i] ? p[i] : 1.0f - p[i];
            acc += __logf(fmaxf(s, NEAR_0));
        }
    } else if (bid < NLL_BLOCKS + W1_BLOCKS) {
        const int gtid = (bid - NLL_BLOCKS) * THREADS + threadIdx.x;
        acc = ssq_partial(W1, nw1, gtid, W1_BLOCKS * THREADS);
    } else {
        const int gtid = (bid - NLL_BLOCKS - W1_BLOCKS) * THREADS + threadIdx.x;
        acc = ssq_partial(W2, nw2, gtid, W2_BLOCKS * THREADS);
    }

    float total = block_sum_wmma(acc);   // all threads reach the barrier
    if (threadIdx.x == 0) partials[bid] = total;
}

__global__ __launch_bounds__(THREADS) void finalize_kernel(
    const float* __restrict__ partials, float* __restrict__ out, double invN)
{
    __shared__ double s0[THREADS], s1[THREADS], s2[THREADS];
    const int tid = threadIdx.x;
    double a = 0.0, b = 0.0, c = 0.0;
    for (int j = tid; j < NLL_BLOCKS; j += THREADS) a += (double)partials[j];
    for (int j = tid; j < W1_BLOCKS;  j += THREADS) b += (double)partials[NLL_BLOCKS + j];
    for (int j = tid; j < W2_BLOCKS;  j += THREADS) c += (double)partials[NLL_BLOCKS + W1_BLOCKS + j];
    s0[tid] = a; s1[tid] = b; s2[tid] = c;
    __syncthreads();
    for (int st = THREADS / 2; st > 0; st >>= 1) {
        if (tid < st) {
            s0[tid] += s0[tid + st];
            s1[tid] += s1[tid + st];
            s2[tid] += s2[tid + st];
        }
        __syncthreads();
    }
    if (tid == 0) {
        double nll = -s0[0] * invN;
        double reg = 0.002 * (sqrt(s1[0]) + sqrt(s2[0]));
        out[0] = (float)(nll + reg);
    }
}

extern "C" void kernel_launch(void* const* d_in, const int* in_sizes, int n_in,
                              void* d_out, int out_size, void* d_ws, size_t ws_size,
                              hipStream_t stream)
{
    const float* p   = (const float*)d_in[0];  // output_1, [N] f32
    const int*   lab = (const int*)d_in[1];    // label,    [N] i32
    const float* W1  = (const float*)d_in[2];  // [1024*4096] f32
    const float* W2  = (const float*)d_in[3];  // [4096*1024] f32
    const int n   = in_sizes[0];
    const int nw1 = in_sizes[2];
    const int nw2 = in_sizes[3];

    float* partials = (float*)d_ws;            // 3072 floats = 12 KB

    partials_kernel<<<TOTAL_BLOCKS, THREADS, 0, stream>>>(
        p, lab, W1, W2, partials, n, nw1, nw2);
    finalize_kernel<<<1, THREADS, 0, stream>>>(
        partials, (float*)d_out, 1.0 / (double)n);
}